// TypedCoords2Volume_74543452389602
// MI455X (gfx1250) — compile-verified
//
#include <hip/hip_runtime.h>
#include <hip/hip_bf16.h>

// ---------------------------------------------------------------------------
// Gaussian splat (TypedCoords2Volume) for MI455X / gfx1250.
//   out[row, ix, iy, iz] += exp(-0.5*((x-ix*res)^2+(y-iy*res)^2+(z-iz*res)^2))
// over the 5x5x5 neighborhood of each atom's home cell.
// Memory-bound: 304 MB zero-fill + ~11M L2-resident fp32 atomics.
// Fast path (stencil fully in-box, the overwhelmingly common case): no
// per-cell predication, z-runs of 5 contiguous atomics fused into a single
// s_clause of global_atomic_add_f32 with immediate offsets.
// ---------------------------------------------------------------------------

// Native CDNA5 fp32 global atomic add (no return -> STOREcnt, RMW in the L2
// atomic units). Device scope: volumes are shared across workgroups.
__device__ __forceinline__ void atomic_add_f32_dev(float* p, float v) {
    asm volatile("global_atomic_add_f32 %0, %1, off scope:SCOPE_DEV"
                 :
                 : "v"(p), "v"(v)
                 : "memory");
}

// Five consecutive fp32 atomic adds (one z-run of the stencil) issued as a
// single hardware clause: one address, immediate byte offsets 0..16.
__device__ __forceinline__ void atomic_add_f32_run5(float* p, float v0, float v1,
                                                    float v2, float v3, float v4) {
    asm volatile(
        "s_clause 0x4\n\t"
        "global_atomic_add_f32 %0, %1, off scope:SCOPE_DEV\n\t"
        "global_atomic_add_f32 %0, %2, off offset:4 scope:SCOPE_DEV\n\t"
        "global_atomic_add_f32 %0, %3, off offset:8 scope:SCOPE_DEV\n\t"
        "global_atomic_add_f32 %0, %4, off offset:12 scope:SCOPE_DEV\n\t"
        "global_atomic_add_f32 %0, %5, off offset:16 scope:SCOPE_DEV"
        :
        : "v"(p), "v"(v0), "v"(v1), "v"(v2), "v"(v3), "v"(v4)
        : "memory");
}

// Streaming zero-fill of the output volumes: 128-bit stores, grid-stride.
__global__ void __launch_bounds__(256)
splat_zero_kernel(float* __restrict__ out, long long n) {
    const long long n4     = n >> 2;
    const long long stride = (long long)gridDim.x * blockDim.x;
    long long i = (long long)blockIdx.x * blockDim.x + threadIdx.x;
    float4* __restrict__ o4 = reinterpret_cast<float4*>(out);
    const float4 z = make_float4(0.f, 0.f, 0.f, 0.f);
    for (long long k = i; k < n4; k += stride) o4[k] = z;
    const long long tail = n & 3;  // out_size is 4-divisible here, but be safe
    if (i < tail) out[n - 1 - i] = 0.f;
}

// One thread per (row, atom). Separable Gaussian: 15 v_exp_f32 per atom,
// then 125 fp32 atomics into the row's box^3 volume.
__global__ void __launch_bounds__(256)
splat_atoms_kernel(const float* __restrict__ coords,    // [N, A, 3]
                   const int*   __restrict__ num_atoms, // [N]
                   const int*   __restrict__ box_p,     // scalar (device)
                   const float* __restrict__ res_p,     // scalar (device)
                   float*       __restrict__ out,       // [N, box^3]
                   int A) {
    const int row  = blockIdx.y;
    const int atom = blockIdx.x * blockDim.x + threadIdx.x;

    const int na = num_atoms[row];          // uniform -> s_load
    if (atom >= A || atom >= na) return;

    const int   box = box_p[0];             // uniform -> s_load
    const float res = res_p[0];

    const float* c = coords + ((long long)row * A + atom) * 3;
    const float x = c[0];                   // one global_load_b96
    const float y = c[1];
    const float z = c[2];

    // Home cell: floor(coord / res), matching the reference exactly.
    const int cx = (int)floorf(x / res);
    const int cy = (int)floorf(y / res);
    const int cz = (int)floorf(z / res);

    // Per-axis weights for offsets -2..+2 (separable Gaussian: 15 exps).
    float wx[5], wy[5], wz[5];
#pragma unroll
    for (int t = 0; t < 5; ++t) {
        const float dx = x - (float)(cx + t - 2) * res;
        const float dy = y - (float)(cy + t - 2) * res;
        const float dz = z - (float)(cz + t - 2) * res;
        wx[t] = __expf(-0.5f * dx * dx);    // v_exp_f32 (TRANS)
        wy[t] = __expf(-0.5f * dy * dy);
        wz[t] = __expf(-0.5f * dz * dz);
    }

    float* __restrict__ vol =
        out + (long long)row * (long long)box * box * box;

    const bool fast = (cx - 2 >= 0) && (cx + 2 < box) &&
                      (cy - 2 >= 0) && (cy + 2 < box) &&
                      (cz - 2 >= 0) && (cz + 2 < box);

    if (fast) {
        // Whole 5x5x5 stencil in-box: no predication, clause-fused z-runs.
        const long long sy = (long long)box;        // y stride (elements)
        const long long sx = sy * box;              // x stride (elements)
        float* pbase = vol + ((long long)(cx - 2) * box + (cy - 2)) * sy + (cz - 2);
#pragma unroll
        for (int i = 0; i < 5; ++i) {
            float* pi = pbase + (long long)i * sx;
#pragma unroll
            for (int j = 0; j < 5; ++j) {
                const float wxy = wx[i] * wy[j];
                atomic_add_f32_run5(pi + (long long)j * sy,
                                    wxy * wz[0], wxy * wz[1], wxy * wz[2],
                                    wxy * wz[3], wxy * wz[4]);
            }
        }
    } else {
        // Boundary atoms: fully predicated per-cell path.
#pragma unroll
        for (int i = 0; i < 5; ++i) {
            const int ix = cx + (i - 2);
            if ((unsigned)ix >= (unsigned)box) continue;
#pragma unroll
            for (int j = 0; j < 5; ++j) {
                const int iy = cy + (j - 2);
                if ((unsigned)iy >= (unsigned)box) continue;
                const float wxy = wx[i] * wy[j];
                float* p = vol + ((long long)ix * box + iy) * (long long)box;
#pragma unroll
                for (int k = 0; k < 5; ++k) {
                    const int iz = cz + (k - 2);
                    if ((unsigned)iz >= (unsigned)box) continue;
                    atomic_add_f32_dev(p + iz, wxy * wz[k]);
                }
            }
        }
    }
}

extern "C" void kernel_launch(void* const* d_in, const int* in_sizes, int n_in,
                              void* d_out, int out_size, void* d_ws, size_t ws_size,
                              hipStream_t stream) {
    const float* coords = (const float*)d_in[0];  // [B, T, 3*A] fp32
    const int*   natoms = (const int*)  d_in[1];  // [B, T] int32
    const int*   boxp   = (const int*)  d_in[2];  // scalar int (device)
    const float* resp   = (const float*)d_in[3];  // scalar fp32 (device)
    float*       out    = (float*)      d_out;    // [B, T, box^3] fp32

    const int N = in_sizes[1];              // B*T rows
    const int A = in_sizes[0] / (3 * N);    // max atoms per row

    const int threads = 256;                // 8 wave32s per block

    // 1) Zero the output volumes (streaming b128 stores).
    {
        long long n  = (long long)out_size;
        long long n4 = n >> 2;
        long long blocks = (n4 + threads - 1) / threads;
        if (blocks < 1) blocks = 1;
        splat_zero_kernel<<<dim3((unsigned)blocks), threads, 0, stream>>>(out, n);
    }

    // 2) Splat all atoms (one thread per atom, grid.y = row).
    {
        dim3 grid((A + threads - 1) / threads, N);
        splat_atoms_kernel<<<grid, threads, 0, stream>>>(coords, natoms, boxp,
                                                         resp, out, A);
    }
}